// ROIAlign_85581518340178
// MI455X (gfx1250) — compile-verified
//
#include <hip/hip_runtime.h>

typedef __attribute__((ext_vector_type(2))) float v2f;
typedef __attribute__((ext_vector_type(8))) float v8f;

namespace {
constexpr int   Cc    = 256;
constexpr int   Hh    = 200;
constexpr int   Ww    = 200;
constexpr int   PH    = 7;
constexpr int   PW    = 7;
constexpr float SCALE = 0.25f;
constexpr int   KTAPS = 112;   // 4 y-taps * 28 x-taps per (roi, ph)
constexpr int   CHUNK = 64;    // channels staged in LDS per pass
}

// One block (128 threads = 4 wave32) handles one (roi, ph) row:
//   out[pw 0..6, c 0..255] = A[16x112] x S[112 x 256]  via v_wmma_f32_16x16x4_f32
__global__ __launch_bounds__(128)
void roialign_wmma_kernel(const float* __restrict__ input,
                          const float* __restrict__ rois,
                          float* __restrict__ out)
{
    __shared__ float A_s[16][116];          // padded stride: conflict-free A-frag reads
    __shared__ float S_s[KTAPS][CHUNK + 2]; // gathered taps for one channel chunk
    __shared__ float wxt_s[28];
    __shared__ float wyt_s[4];
    __shared__ int   xls_s[14];             // clamped x-lo (<= W-2) per x-sample
    __shared__ int   xedge_s[14];           // 1 if lo==hi==W-1
    __shared__ int   ytap_s[4];             // y row index per y-tap

    const int bid  = blockIdx.x;
    const int n    = bid / PH;
    const int ph   = bid % PH;
    const int t    = threadIdx.x;
    const int lane = t & 31;
    const int wave = t >> 5;

    // ROI parameters (uniform -> scalar loads)
    const float fb = rois[n * 5 + 0];
    const float x1 = rois[n * 5 + 1] * SCALE - 0.5f;
    const float y1 = rois[n * 5 + 2] * SCALE - 0.5f;
    const float x2 = rois[n * 5 + 3] * SCALE - 0.5f;
    const float y2 = rois[n * 5 + 4] * SCALE - 0.5f;
    const int   b  = (int)fb;
    const float bin_w = (x2 - x1) * (1.0f / PW);
    const float bin_h = (y2 - y1) * (1.0f / PH);

    // ---- per-axis sample metadata (reference _axis_samples semantics) ----
    if (t < 14) {                       // x samples: j = pw*2 + s
        const int j = t;
        float coord = x1 + (float)(j >> 1) * bin_w + ((float)(j & 1) + 0.5f) * bin_w * 0.5f;
        bool  valid = (coord >= -1.0f) && (coord <= (float)Ww);
        float c     = fmaxf(coord, 0.0f);
        int   lo_raw = (int)floorf(c);
        bool  at_edge = lo_raw >= (Ww - 1);
        int   lo   = at_edge ? (Ww - 1) : lo_raw;
        float frac = at_edge ? 0.0f : (c - (float)lo);
        wxt_s[2 * j]     = valid ? (1.0f - frac) : 0.0f;
        wxt_s[2 * j + 1] = valid ? frac : 0.0f;
        xls_s[j]   = at_edge ? (Ww - 2) : lo;
        xedge_s[j] = at_edge ? 1 : 0;
    } else if (t >= 32 && t < 34) {     // y samples for this ph: s = 0,1
        const int s = t - 32;
        float coord = y1 + (float)ph * bin_h + ((float)s + 0.5f) * bin_h * 0.5f;
        bool  valid = (coord >= -1.0f) && (coord <= (float)Hh);
        float c     = fmaxf(coord, 0.0f);
        int   lo_raw = (int)floorf(c);
        bool  at_edge = lo_raw >= (Hh - 1);
        int   lo   = at_edge ? (Hh - 1) : lo_raw;
        float frac = at_edge ? 0.0f : (c - (float)lo);
        ytap_s[2 * s]     = lo;
        ytap_s[2 * s + 1] = at_edge ? (Hh - 1) : (lo + 1);
        wyt_s[2 * s]      = valid ? (1.0f - frac) : 0.0f;
        wyt_s[2 * s + 1]  = valid ? frac : 0.0f;
    }
    __syncthreads();

    // ---- build A[16x112]: A[pw][iy*28+m] = 0.25*wy[iy]*wx[m] iff m>>2 == pw ----
    for (int i = t; i < 16 * KTAPS; i += 128) {
        const int col = i >> 4;
        const int row = i & 15;
        const int iy  = col / 28;
        const int m   = col - iy * 28;
        float v = 0.0f;
        if ((m >> 2) == row)            // rows 7..15 never match (m>>2 <= 6)
            v = 0.25f * wyt_s[iy] * wxt_s[m];
        A_s[row][col] = v;
    }
    __syncthreads();

    const size_t in_base_b = (size_t)b * Cc * Hh * Ww;

    for (int c0 = 0; c0 < Cc; c0 += CHUNK) {
        // ---- gather taps for this channel chunk into LDS ----
        // lanes 0..27 fetch the 28 x-taps of one (y-row, channel): addresses
        // stay within one 800B image row -> good cacheline locality.
        for (int it = wave; it < 4 * CHUNK; it += 4) {
            const int iy = it & 3;
            const int cl = it >> 2;
            if (lane < 28) {
                const int j = lane >> 1;
                const int x = xls_s[j] + ((lane & 1) | xedge_s[j]);
                const int y = ytap_s[iy];
                const float v =
                    input[in_base_b + ((size_t)(c0 + cl) * Hh + y) * Ww + x];
                S_s[iy * 28 + lane][cl] = v;
            }
        }
        __syncthreads();

        // ---- WMMA: each wave owns one 16-channel N-tile, K=112 in steps of 4 ----
        v8f acc = {};
        const int arow = lane & 15;
        const int koff = (lane >> 4) << 1;          // lanes 0-15 -> K{0,1}; 16-31 -> K{2,3}
        const int bn   = (wave << 4) + (lane & 15); // channel within chunk
        for (int kk = 0; kk < 28; ++kk) {
            v2f a, bb;
            a.x  = A_s[arow][kk * 4 + koff];
            a.y  = A_s[arow][kk * 4 + koff + 1];
            bb.x = S_s[kk * 4 + koff][bn];
            bb.y = S_s[kk * 4 + koff + 1][bn];
            acc = __builtin_amdgcn_wmma_f32_16x16x4_f32(
                /*neg_a=*/false, a, /*neg_b=*/false, bb,
                /*c_mod=*/(short)0, acc, /*reuse_a=*/false, /*reuse_b=*/false);
        }

        // ---- store: D rows 0..6 = pw (lanes 0-15 hold M=0..7 in v[0..7]) ----
        if (lane < 16) {
            const int c     = c0 + (wave << 4) + lane;
            const int obase = ((n * Cc + c) * PH + ph) * PW;
#pragma unroll
            for (int r = 0; r < PW; ++r)
                out[obase + r] = acc[r];
        }
        __syncthreads();   // before next chunk overwrites S_s
    }
}

extern "C" void kernel_launch(void* const* d_in, const int* in_sizes, int n_in,
                              void* d_out, int out_size, void* d_ws, size_t ws_size,
                              hipStream_t stream) {
    const float* input = (const float*)d_in[0];
    const float* rois  = (const float*)d_in[1];
    float*       outp  = (float*)d_out;
    const int n_rois = in_sizes[1] / 5;

    dim3 grid(n_rois * PH);
    dim3 block(128);
    roialign_wmma_kernel<<<grid, block, 0, stream>>>(input, rois, outp);
}